// FHNNLayer_26680336843173
// MI455X (gfx1250) — compile-verified
//
#include <hip/hip_runtime.h>
#include <math.h>

#define C_CURV   0.01f
#define SQRT_C   0.1f       // sqrt(0.01)
#define EPS_W    1e-6f
#define MIN_NORM 1e-10f
#define BALL_EPS 1e-5f

#define RMAX     230        // relation count in the reference workload
#define WSTRIDE  257        // padded LDS stride (257 % 64 == 1 -> bank de-conflict)

typedef __attribute__((ext_vector_type(2))) float v2f;
typedef __attribute__((ext_vector_type(8))) float v8f;
typedef __attribute__((ext_vector_type(4))) unsigned int v4u;
typedef __attribute__((ext_vector_type(8))) int v8i;
typedef __attribute__((ext_vector_type(4))) int v4i;

__device__ __forceinline__ float sq4(float4 a) {
    return a.x*a.x + a.y*a.y + a.z*a.z + a.w*a.w;
}

// Raw LDS byte offset of a __shared__ object (generic -> addrspace(3) -> int).
__device__ __forceinline__ unsigned lds_offset_bytes(const void* p) {
    return (unsigned)(size_t)(const __attribute__((address_space(3))) void*)p;
}

// ---------------------------------------------------------------- zero ws
__global__ void zero_kernel(float* __restrict__ p, int n) {
    int i = blockIdx.x * blockDim.x + threadIdx.x;
    if (i < n) p[i] = 0.0f;
}

// ---------------------------------------------------------------- log map
__global__ void log_map_kernel(const float* __restrict__ x,
                               float* __restrict__ h_tan, int N) {
    int i = blockIdx.x * blockDim.x + threadIdx.x;
    if (i >= N) return;
    const float4* xr = (const float4*)(x + (size_t)i * 16);
    float4 a = xr[0], b = xr[1], c = xr[2], d = xr[3];
    float ss = sq4(a) + sq4(b) + sq4(c) + sq4(d);
    float n  = sqrtf(fmaxf(ss, MIN_NORM));
    float scn = SQRT_C * n;
    float cl  = fminf(fmaxf(scn, -1.0f + BALL_EPS), 1.0f - BALL_EPS);
    float f   = atanhf(cl) / scn;
    float4* o = (float4*)(h_tan + (size_t)i * 16);
    a.x*=f; a.y*=f; a.z*=f; a.w*=f;  b.x*=f; b.y*=f; b.z*=f; b.w*=f;
    c.x*=f; c.y*=f; c.z*=f; c.w*=f;  d.x*=f; d.y*=f; d.z*=f; d.w*=f;
    o[0]=a; o[1]=b; o[2]=c; o[3]=d;
}

// ---------------------------------------------------------------- loop GEMM (WMMA fp32)
// loop_tan[16-row tile] = h_tan_tile(16x16) @ W(16x16), K=16 as 4 x V_WMMA_F32_16X16X4_F32
__global__ __launch_bounds__(256)
void loop_gemm_kernel(const float* __restrict__ h_tan,
                      const float* __restrict__ W,       // 16x16 row-major
                      float* __restrict__ loop_tan, int N) {
    __shared__ float sW[256];
    sW[threadIdx.x] = W[threadIdx.x];          // blockDim == 256 exactly
    __syncthreads();

    int wave = (blockIdx.x * blockDim.x + threadIdx.x) >> 5;   // global wave id
    int lane = threadIdx.x & 31;
    int half = lane >> 4;                       // 0: lanes 0-15, 1: lanes 16-31
    int m    = lane & 15;
    int base = wave * 16;
    if (base >= N) return;                      // uniform per wave -> EXEC all-ones at WMMA

    int row = base + m; if (row >= N) row = N - 1;
    const float4* hr = (const float4*)(h_tan + (size_t)row * 16);
    float4 h0 = hr[0], h1 = hr[1], h2 = hr[2], h3 = hr[3];

    v8f acc = {};   // C = 0
    // A (ISA 7.12.2, 32-bit A 16x4): lanes 0-15 hold K=0(v0),K=1(v1); lanes 16-31 K=2,K=3
    // B (4x16): v0 = row K=2*half, v1 = row K=2*half+1, N = lane&15
    {
        v2f A, B; int k;
        k = 0 + 2*half;
        A.x = half ? h0.z : h0.x;  A.y = half ? h0.w : h0.y;
        B.x = sW[k*16 + m];        B.y = sW[(k+1)*16 + m];
        acc = __builtin_amdgcn_wmma_f32_16x16x4_f32(false, A, false, B, (short)0, acc, false, false);
        k = 4 + 2*half;
        A.x = half ? h1.z : h1.x;  A.y = half ? h1.w : h1.y;
        B.x = sW[k*16 + m];        B.y = sW[(k+1)*16 + m];
        acc = __builtin_amdgcn_wmma_f32_16x16x4_f32(false, A, false, B, (short)0, acc, false, false);
        k = 8 + 2*half;
        A.x = half ? h2.z : h2.x;  A.y = half ? h2.w : h2.y;
        B.x = sW[k*16 + m];        B.y = sW[(k+1)*16 + m];
        acc = __builtin_amdgcn_wmma_f32_16x16x4_f32(false, A, false, B, (short)0, acc, false, false);
        k = 12 + 2*half;
        A.x = half ? h3.z : h3.x;  A.y = half ? h3.w : h3.y;
        B.x = sW[k*16 + m];        B.y = sW[(k+1)*16 + m];
        acc = __builtin_amdgcn_wmma_f32_16x16x4_f32(false, A, false, B, (short)0, acc, false, false);
    }
    // D layout: VGPR j holds element (M = j + 8*half, N = m)
    if (base + 16 <= N) {                       // wave-uniform fast path: no per-store guards
        #pragma unroll
        for (int j = 0; j < 8; ++j)
            loop_tan[(size_t)(base + j + 8*half) * 16 + m] = acc[j];
    } else {
        #pragma unroll
        for (int j = 0; j < 8; ++j) {
            int r = base + j + 8*half;
            if (r < N) loop_tan[(size_t)r * 16 + m] = acc[j];
        }
    }
}

// ---------------------------------------------------------------- norm_sum (degree-weighted)
__global__ void degree_kernel(const float* __restrict__ node_norm,
                              const int* __restrict__ dst,
                              float* __restrict__ norm_sum, int E) {
    int e = blockIdx.x * blockDim.x + threadIdx.x;
    if (e >= E) return;
    int t = dst[e];
    atomicAdd(&norm_sum[t], node_norm[t]);
}

// ---------------------------------------------------------------- edge math (shared by both paths)
__device__ __forceinline__ void edge_compute(int o,
                                             const float* __restrict__ h_tan,
                                             const float* __restrict__ Wrow,  // base of W[r]
                                             float emb,
                                             const float* __restrict__ node_norm,
                                             const float* __restrict__ norm_sum,
                                             int s, int t,
                                             float* __restrict__ numer,
                                             float* __restrict__ denom) {
    float vo = h_tan[(size_t)s * 16 + o];
    float acc = emb;
    #pragma unroll
    for (int k = 0; k < 16; ++k) {
        float vk = __shfl(vo, k, 16);               // broadcast v[k] within half-wave
        acc += vk * Wrow[k * 16 + o];
    }
    float ss = acc * acc;                            // |msg_tan|^2 across 16 lanes
    #pragma unroll
    for (int msk = 1; msk < 16; msk <<= 1) ss += __shfl_xor(ss, msk, 16);

    float n   = sqrtf(fmaxf(ss, MIN_NORM));
    float scn = SQRT_C * n;
    float f   = tanhf(scn) / scn;                    // exp_map_zero factor
    float mo  = f * acc;                             // msg[o]
    float msq = f * f * ss;                          // |msg|^2 (analytic)
    float lam = 2.0f / (1.0f - C_CURV * msq + EPS_W);
    float w   = node_norm[t] / (norm_sum[t] + EPS_W);
    float wl  = w * lam;
    atomicAdd(&numer[(size_t)t * 16 + o], wl * mo);
    if (o == 0) atomicAdd(&denom[t], wl);
}

// ---------------------------------------------------------------- edge kernel, LDS-staged W
// Persistent blocks: each block stages ALL R relation matrices (+rel_emb) into the
// 320KB-per-workgroup LDS once, then grid-strides over edges. W reads hit LDS, not L2.
// Staging of rel_weight uses the Tensor Data Mover: one D# with pad_enable reproduces
// the 257-float bank-de-conflict stride (pad 1 DWORD after every 256 DWORDs) in HW.
__global__ __launch_bounds__(1024)
void edge_kernel_lds(const float* __restrict__ h_tan,
                     const float* __restrict__ rel_w,    // [R,16,16]
                     const float* __restrict__ rel_emb,  // [R,16]
                     const float* __restrict__ node_norm,
                     const float* __restrict__ norm_sum,
                     const int* __restrict__ src,
                     const int* __restrict__ dst,
                     const int* __restrict__ ety,
                     float* __restrict__ numer,
                     float* __restrict__ denom, int E, int R) {
    __shared__ float sW[RMAX * WSTRIDE];   // padded stride: half-waves land on disjoint banks
    __shared__ float sE[RMAX * 16];

#if __has_builtin(__builtin_amdgcn_tensor_load_to_lds)
    if (threadIdx.x < 32) {                // wave 0 issues the TDM op (EXEC-independent)
        unsigned nelem = (unsigned)(R * 256);            // elements, data_size = 4B
        unsigned lds0  = lds_offset_bytes(sW);
        unsigned long long ga = (unsigned long long)(size_t)rel_w;
        v4u g0; v8i g1; v4i g2 = {0,0,0,0}, g3 = {0,0,0,0};
        v8i g4 = {0,0,0,0,0,0,0,0};                      // extra group (6-arg toolchain), unused
        // ---- D# group 0: count=1 | lds_addr | global_addr[56:0] | type=2
        g0[0] = 1u;                                      // count = 1 valid descriptor
        g0[1] = lds0;                                    // LDS byte address of sW
        g0[2] = (unsigned)ga;                            // global_addr[31:0]
        g0[3] = ((unsigned)(ga >> 32) & 0x01FFFFFFu) | (2u << 30);  // addr[56:32] | type=2
        // ---- D# group 1: data_size=4B, pad after every 256 DWORDs by 1 DWORD
        g1[0] = (int)((2u << 16)        // data_size = 2 (4 bytes)
                    | (1u << 20)        // pad_enable
                    | (7u << 22));      // pad_interval = 7 (256 DWORDs); pad_amount = 0 (1 DWORD)
        g1[1] = (int)((nelem & 0xFFFFu) << 16);          // tensor_dim0[15:0]  @bits[63:48]
        g1[2] = (int)(((nelem >> 16) & 0xFFFFu)          // tensor_dim0[31:16] @bits[79:64]
                    | (1u << 16));                       // tensor_dim1 = 1    @bits[111:80]
        g1[3] = (int)((nelem & 0xFFFFu) << 16);          // tile_dim0 = nelem  @bits[127:112]
        g1[4] = 0;                                       // tile_dim1 = tile_dim2 = 0 (unused)
        g1[5] = (int)nelem;                              // tensor_dim0_stride[31:0]
        g1[6] = 0; g1[7] = 0;
        __builtin_amdgcn_tensor_load_to_lds(g0, g1, g2, g3, g4, 0);
        __builtin_amdgcn_s_wait_tensorcnt(0);            // TENSORcnt -> 0 before barrier
    }
#else
    for (int i = threadIdx.x; i < R * 256; i += blockDim.x) {
        int r = i >> 8, off = i & 255;
        sW[r * WSTRIDE + off] = rel_w[i];                // fallback: load/store staging
    }
#endif
    for (int i = threadIdx.x; i < R * 16; i += blockDim.x)
        sE[i] = rel_emb[i];
    __syncthreads();

    int o        = threadIdx.x & 15;
    int hw_local = threadIdx.x >> 4;                   // half-wave id in block
    int hw_per_b = blockDim.x >> 4;
    for (int e = blockIdx.x * hw_per_b + hw_local; e < E; e += gridDim.x * hw_per_b) {
        int s = src[e], t = dst[e], r = ety[e];
        edge_compute(o, h_tan, sW + (size_t)r * WSTRIDE, sE[r * 16 + o],
                     node_norm, norm_sum, s, t, numer, denom);
    }
}

// ---------------------------------------------------------------- edge kernel, global-W fallback
__global__ void edge_kernel_glb(const float* __restrict__ h_tan,
                                const float* __restrict__ rel_w,
                                const float* __restrict__ rel_emb,
                                const float* __restrict__ node_norm,
                                const float* __restrict__ norm_sum,
                                const int* __restrict__ src,
                                const int* __restrict__ dst,
                                const int* __restrict__ ety,
                                float* __restrict__ numer,
                                float* __restrict__ denom, int E) {
    int gid = blockIdx.x * blockDim.x + threadIdx.x;
    int e = gid >> 4;
    int o = gid & 15;
    if (e >= E) return;
    int s = src[e], t = dst[e], r = ety[e];
    edge_compute(o, h_tan, rel_w + (size_t)r * 256, rel_emb[(size_t)r * 16 + o],
                 node_norm, norm_sum, s, t, numer, denom);
}

// ---------------------------------------------------------------- final fuse per node
__global__ void final_kernel(const float* __restrict__ numer,
                             const float* __restrict__ denom,
                             const float* __restrict__ loop_tan,
                             float* __restrict__ out, int N) {
    int i = blockIdx.x * blockDim.x + threadIdx.x;
    if (i >= N) return;
    float xv[16], yv[16];
    const float* nr = numer + (size_t)i * 16;
    const float* lr = loop_tan + (size_t)i * 16;
    float dinv = 1.0f / (denom[i] + EPS_W);
    float ssx = 0.0f, ssl = 0.0f;
    #pragma unroll
    for (int j = 0; j < 16; ++j) {
        xv[j] = nr[j] * dinv;  ssx += xv[j] * xv[j];
        yv[j] = lr[j];         ssl += yv[j] * yv[j];
    }
    // project_to_ball(h_agg)
    float nx   = sqrtf(fmaxf(ssx, MIN_NORM));
    float maxn = (1.0f - BALL_EPS) / SQRT_C;
    if (nx > maxn) {
        float sc = maxn / nx;
        ssx = 0.0f;
        #pragma unroll
        for (int j = 0; j < 16; ++j) { xv[j] *= sc; ssx += xv[j] * xv[j]; }
    }
    // loop_hyp = exp_map_zero(loop_tan)
    float nl  = sqrtf(fmaxf(ssl, MIN_NORM));
    float scn = SQRT_C * nl;
    float fl  = tanhf(scn) / scn;
    float ssy = 0.0f, xy = 0.0f;
    #pragma unroll
    for (int j = 0; j < 16; ++j) {
        yv[j] *= fl;  ssy += yv[j] * yv[j];  xy += xv[j] * yv[j];
    }
    // mobius_add(x, y)
    float a   = 1.0f + 2.0f * C_CURV * xy + C_CURV * ssy;
    float b   = 1.0f - C_CURV * ssx;
    float den = 1.0f + 2.0f * C_CURV * xy + C_CURV * C_CURV * ssx * ssy;
    float di  = 1.0f / fmaxf(den, MIN_NORM);
    float* orow = out + (size_t)i * 16;
    #pragma unroll
    for (int j = 0; j < 16; ++j) orow[j] = (a * xv[j] + b * yv[j]) * di;
}

// ================================================================ launcher
extern "C" void kernel_launch(void* const* d_in, const int* in_sizes, int n_in,
                              void* d_out, int out_size, void* d_ws, size_t ws_size,
                              hipStream_t stream) {
    const float* h_hyper   = (const float*)d_in[0];
    const float* node_norm = (const float*)d_in[1];
    const float* rel_w     = (const float*)d_in[2];
    const float* loop_w    = (const float*)d_in[3];
    const float* rel_emb   = (const float*)d_in[4];
    const int*   src       = (const int*)d_in[5];
    const int*   dst       = (const int*)d_in[6];
    const int*   ety       = (const int*)d_in[7];

    const int N = in_sizes[0] / 16;
    const int E = in_sizes[5];
    const int R = in_sizes[4] / 16;

    // workspace layout (floats): [norm_sum N][denom N][numer 16N][h_tan 16N][loop_tan 16N]
    float* ws       = (float*)d_ws;
    float* norm_sum = ws;
    float* denom    = ws + (size_t)N;
    float* numer    = ws + (size_t)2  * N;
    float* h_tan    = ws + (size_t)18 * N;
    float* loop_tan = ws + (size_t)34 * N;

    const int B = 256;
    // 1) zero accumulators (norm_sum + denom + numer = 18N floats), every call
    {
        int n = 18 * N;
        zero_kernel<<<(n + B - 1) / B, B, 0, stream>>>(ws, n);
    }
    // 2) log map
    log_map_kernel<<<(N + B - 1) / B, B, 0, stream>>>(h_hyper, h_tan, N);
    // 3) loop GEMM via fp32 WMMA (16 rows per wave, 8 waves per block)
    {
        int waves  = (N + 15) / 16;
        int blocks = (waves + 7) / 8;
        loop_gemm_kernel<<<blocks, 256, 0, stream>>>(h_tan, loop_w, loop_tan, N);
    }
    // 4) norm_sum segment reduction
    degree_kernel<<<(E + B - 1) / B, B, 0, stream>>>(node_norm, dst, norm_sum, E);
    // 5) per-edge message + midpoint accumulation (16 lanes per edge)
    if (R <= RMAX) {
        int hw_per_b = 1024 >> 4;                       // 64 half-waves per block
        int blocks   = 128;                             // persistent blocks; ~32MB staging total
        int maxb     = (E + hw_per_b - 1) / hw_per_b;
        if (blocks > maxb) blocks = maxb;
        edge_kernel_lds<<<blocks, 1024, 0, stream>>>(h_tan, rel_w, rel_emb, node_norm,
                                                     norm_sum, src, dst, ety,
                                                     numer, denom, E, R);
    } else {
        long long thr = (long long)E * 16;
        int blocks = (int)((thr + B - 1) / B);
        edge_kernel_glb<<<blocks, B, 0, stream>>>(h_tan, rel_w, rel_emb, node_norm,
                                                  norm_sum, src, dst, ety,
                                                  numer, denom, E);
    }
    // 6) final per-node fuse
    final_kernel<<<(N + B - 1) / B, B, 0, stream>>>(numer, denom, loop_tan, (float*)d_out, N);
}